// OpenPanguMoEDecoderLayer_9620726743830
// MI455X (gfx1250) — compile-verified
//
#include <hip/hip_runtime.h>
#include <hip/hip_bf16.h>
#include <math.h>

// ---------------------------------------------------------------------------
// MoE (OpenPangu-style): top-2 of 16 routed experts + shared expert, SwiGLU.
// T=1024 tokens, H=1024, M=512, E=16, fp32 everywhere (matches reference).
// GEMM engine: V_WMMA_F32_16X16X4_F32 (fp32-exact CDNA5 matrix pipe).
// Each wave owns TWO 16x16 output tiles -> independent WMMA accumulator
// chains (pipelines the XDL) and 2x A-fragment reuse per LDS read.
// ---------------------------------------------------------------------------

#define TT 1024   // tokens
#define HH 1024   // hidden
#define MM 512    // moe intermediate
#define EE 16     // experts
#define ROUTED_SCALE 2.5f

typedef float v2f __attribute__((ext_vector_type(2)));
typedef float v8f __attribute__((ext_vector_type(8)));

__device__ __forceinline__ v8f wmma4(v2f a, v2f b, v8f c) {
  // D = A(16x4 f32) * B(4x16 f32) + C(16x16 f32)
  return __builtin_amdgcn_wmma_f32_16x16x4_f32(
      /*neg_a=*/false, a, /*neg_b=*/false, b,
      /*c_mod=*/(short)0, c, /*reuse_a=*/false, /*reuse_b=*/false);
}

// ---------------------------------------------------------------------------
// 1) Gating: scores = sigmoid(x @ gate_w^T); top-2; normalize; *2.5
//    block = 256 threads = 16 experts x 16 lanes; one block per token.
// ---------------------------------------------------------------------------
__global__ __launch_bounds__(256) void gate_kernel(
    const float* __restrict__ X, const float* __restrict__ GW,
    int* __restrict__ counts, int* __restrict__ eids, float* __restrict__ ew) {
  const int t = blockIdx.x;
  const int tid = threadIdx.x;
  const int e = tid >> 4;
  const int l = tid & 15;

  const float* x = X + (size_t)t * HH;
  const float* g = GW + (size_t)e * HH;
  float s = 0.f;
  for (int i = l; i < HH; i += 16) s += x[i] * g[i];
  // butterfly reduce within the 16-lane group (wave32: masks < 16 stay in-half)
  s += __shfl_xor(s, 8);
  s += __shfl_xor(s, 4);
  s += __shfl_xor(s, 2);
  s += __shfl_xor(s, 1);

  __shared__ float sc[EE];
  if (l == 0) sc[e] = s;
  __syncthreads();

  if (tid == 0) {
    float p[EE];
#pragma unroll
    for (int i = 0; i < EE; i++) p[i] = 1.f / (1.f + __expf(-sc[i]));
    // top-2, first-occurrence tie-break (matches lax.top_k)
    int i1 = 0; float m1 = p[0];
#pragma unroll
    for (int i = 1; i < EE; i++) if (p[i] > m1) { m1 = p[i]; i1 = i; }
    int i2 = -1; float m2 = -1.f;
#pragma unroll
    for (int i = 0; i < EE; i++) if (i != i1 && p[i] > m2) { m2 = p[i]; i2 = i; }
    const float inv = 1.f / (m1 + m2 + 1e-20f);
    eids[2 * t]     = i1;  ew[2 * t]     = m1 * inv * ROUTED_SCALE;
    eids[2 * t + 1] = i2;  ew[2 * t + 1] = m2 * inv * ROUTED_SCALE;
    atomicAdd(&counts[i1], 1);
    atomicAdd(&counts[i2], 1);
  }
}

// ---------------------------------------------------------------------------
// 2) Exclusive scan of 16 expert counts.
// ---------------------------------------------------------------------------
__global__ void scan_kernel(const int* __restrict__ counts, int* __restrict__ offsets) {
  if (threadIdx.x == 0 && blockIdx.x == 0) {
    int o = 0;
#pragma unroll
    for (int e = 0; e < EE; e++) { offsets[e] = o; o += counts[e]; }
  }
}

// ---------------------------------------------------------------------------
// 3) Deterministic compaction: slot = offset[e] + rank (tokens kept in order).
// ---------------------------------------------------------------------------
__global__ __launch_bounds__(256) void fill_kernel(
    const int* __restrict__ eids, const float* __restrict__ ew,
    const int* __restrict__ offsets, int* __restrict__ toklist,
    float* __restrict__ slotw) {
  const int t = blockIdx.x * 256 + threadIdx.x;
  if (t >= TT) return;
  const int e0 = eids[2 * t], e1 = eids[2 * t + 1];
  int r0 = 0, r1 = 0;
  for (int tp = 0; tp < t; tp++) {
    const int a = eids[2 * tp], b = eids[2 * tp + 1];
    r0 += (a == e0) + (b == e0);
    r1 += (a == e1) + (b == e1);
  }
  const int s0 = offsets[e0] + r0, s1 = offsets[e1] + r1;
  toklist[s0] = t; slotw[s0] = ew[2 * t];
  toklist[s1] = t; slotw[s1] = ew[2 * t + 1];
}

// ---------------------------------------------------------------------------
// A-tile stage: 16 rows x 64 K floats, float4-vectorized global loads.
// LDS row stride 68 floats (272 B, 16B-aligned -> float4 stores legal;
// 4*row+k bank pattern kills conflicts on frag reads).
// ---------------------------------------------------------------------------
__device__ __forceinline__ void stage_a_tile(
    float (*As)[68], const float* __restrict__ src_base, size_t row_stride,
    const int* __restrict__ toklist, int base, int tile, int count,
    int is_direct, int kc, int tid) {
  for (int j = tid; j < 16 * 16; j += 128) {
    const int r = j >> 4, c4 = j & 15;
    int slot = tile * 16 + r; if (slot >= count) slot = count - 1;
    const int row = is_direct ? (base + slot)
                              : (toklist ? toklist[base + slot] : base + slot);
    const float4 v =
        *(const float4*)(src_base + (size_t)row * row_stride + kc + 4 * c4);
    *(float4*)&As[r][4 * c4] = v;
  }
}

// ---------------------------------------------------------------------------
// 4) act = silu(x@Wg) * (x@Wu), grouped per expert (or shared expert).
//    Block: 128 thr = 4 waves; tile = 16 tokens x 128 M-cols (32 per wave).
//    4 independent WMMA chains per wave (g0,u0,g1,u1).
// ---------------------------------------------------------------------------
__global__ __launch_bounds__(128) void act_kernel(
    const float* __restrict__ X, const float* __restrict__ Wg,
    const float* __restrict__ Wu, const int* __restrict__ toklist,
    const int* __restrict__ counts, const int* __restrict__ offsets,
    float* __restrict__ act, int is_shared) {
  const int e = blockIdx.z;
  int base, count;
  if (is_shared) { base = 0; count = TT; }
  else           { count = counts[e]; base = offsets[e]; }
  const int tile = blockIdx.x;
  if (tile * 16 >= count) return;

  const int tid  = threadIdx.x;
  const int wave = tid >> 5, lane = tid & 31;
  const int row16 = lane & 15;
  const int kh = (lane >> 4) * 2;                         // K-half select
  const int mcol0 = blockIdx.y * 128 + wave * 32 + row16; // tile 0 column
  const int mcol1 = mcol0 + 16;                           // tile 1 column

  const size_t wbase = is_shared ? 0 : (size_t)e * HH * MM;
  const float* wg = Wg + wbase;
  const float* wu = Wu + wbase;

  __shared__ float As[16][68];
  v8f accg0 = {}, accu0 = {}, accg1 = {}, accu1 = {};

  for (int kc = 0; kc < HH; kc += 64) {
    stage_a_tile(As, X, HH, toklist, base, tile, count, is_shared, kc, tid);
    __syncthreads();
#pragma unroll
    for (int k0 = 0; k0 < 64; k0 += 4) {
      v2f a; a.x = As[row16][k0 + kh]; a.y = As[row16][k0 + kh + 1];
      const size_t bi0 = (size_t)(kc + k0 + kh) * MM + mcol0;
      v2f bg0; bg0.x = wg[bi0];      bg0.y = wg[bi0 + MM];
      v2f bu0; bu0.x = wu[bi0];      bu0.y = wu[bi0 + MM];
      v2f bg1; bg1.x = wg[bi0 + 16]; bg1.y = wg[bi0 + MM + 16];
      v2f bu1; bu1.x = wu[bi0 + 16]; bu1.y = wu[bi0 + MM + 16];
      accg0 = wmma4(a, bg0, accg0);
      accu0 = wmma4(a, bu0, accu0);
      accg1 = wmma4(a, bg1, accg1);
      accu1 = wmma4(a, bu1, accu1);
    }
    __syncthreads();
  }
#pragma unroll
  for (int v = 0; v < 8; v++) {
    const int r = v + 8 * (lane >> 4);
    const int slot = tile * 16 + r;
    if (slot < count) {
      float* dst = act + (size_t)(base + slot) * MM;
      const float g0 = accg0[v], u0 = accu0[v];
      const float g1 = accg1[v], u1 = accu1[v];
      dst[mcol0] = (g0 / (1.f + __expf(-g0))) * u0;  // silu(g)*u
      dst[mcol1] = (g1 / (1.f + __expf(-g1))) * u1;
    }
  }
}

// ---------------------------------------------------------------------------
// 5) down: y = act @ Wd; shared pass -> plain store; routed -> w * atomicAdd.
//    Block: 128 thr; tile = 16 tokens x 128 H-cols (32 per wave, 2 chains).
// ---------------------------------------------------------------------------
__global__ __launch_bounds__(128) void down_kernel(
    const float* __restrict__ act, const float* __restrict__ Wd,
    const int* __restrict__ toklist, const float* __restrict__ slotw,
    const int* __restrict__ counts, const int* __restrict__ offsets,
    float* __restrict__ out, int is_shared) {
  const int e = blockIdx.z;
  int base, count;
  if (is_shared) { base = 0; count = TT; }
  else           { count = counts[e]; base = offsets[e]; }
  const int tile = blockIdx.x;
  if (tile * 16 >= count) return;

  const int tid  = threadIdx.x;
  const int wave = tid >> 5, lane = tid & 31;
  const int row16 = lane & 15;
  const int kh = (lane >> 4) * 2;
  const int hcol0 = blockIdx.y * 128 + wave * 32 + row16;
  const int hcol1 = hcol0 + 16;

  const float* wd = Wd + (is_shared ? 0 : (size_t)e * MM * HH);

  __shared__ float As[16][68];
  v8f acc0 = {}, acc1 = {};

  for (int kc = 0; kc < MM; kc += 64) {
    // A rows are contiguous slots of the act buffer (no gather).
    stage_a_tile(As, act, MM, nullptr, base, tile, count, 1, kc, tid);
    __syncthreads();
#pragma unroll
    for (int k0 = 0; k0 < 64; k0 += 4) {
      v2f a; a.x = As[row16][k0 + kh]; a.y = As[row16][k0 + kh + 1];
      const size_t bi0 = (size_t)(kc + k0 + kh) * HH + hcol0;
      v2f b0; b0.x = wd[bi0];      b0.y = wd[bi0 + HH];
      v2f b1; b1.x = wd[bi0 + 16]; b1.y = wd[bi0 + HH + 16];
      acc0 = wmma4(a, b0, acc0);
      acc1 = wmma4(a, b1, acc1);
    }
    __syncthreads();
  }
#pragma unroll
  for (int v = 0; v < 8; v++) {
    const int r = v + 8 * (lane >> 4);
    const int slot = tile * 16 + r;
    if (slot < count) {
      if (is_shared) {
        float* dst = out + (size_t)(base + slot) * HH;
        dst[hcol0] = acc0[v];
        dst[hcol1] = acc1[v];
      } else {
        const int tok = toklist[base + slot];
        const float w = slotw[base + slot];
        atomicAdd(&out[(size_t)tok * HH + hcol0], w * acc0[v]);
        atomicAdd(&out[(size_t)tok * HH + hcol1], w * acc1[v]);
      }
    }
  }
}

// ---------------------------------------------------------------------------
// Host side
// ---------------------------------------------------------------------------
extern "C" void kernel_launch(void* const* d_in, const int* in_sizes, int n_in,
                              void* d_out, int out_size, void* d_ws, size_t ws_size,
                              hipStream_t stream) {
  const float* X   = (const float*)d_in[0];  // [1,1024,1024]
  const float* GW  = (const float*)d_in[1];  // [16,1024]
  const float* Wg  = (const float*)d_in[2];  // [16,1024,512]
  const float* Wu  = (const float*)d_in[3];  // [16,1024,512]
  const float* Wd  = (const float*)d_in[4];  // [16,512,1024]
  const float* WSg = (const float*)d_in[5];  // [1024,512]
  const float* WSu = (const float*)d_in[6];  // [1024,512]
  const float* WSd = (const float*)d_in[7];  // [512,1024]
  float* out = (float*)d_out;                // [1,1024,1024]

  char* ws = (char*)d_ws;
  int*   counts  = (int*)(ws + 0);        // 16 i32
  int*   offsets = (int*)(ws + 64);       // 16 i32
  int*   eids    = (int*)(ws + 256);      // 2048 i32
  float* ew      = (float*)(ws + 8448);   // 2048 f32
  int*   toklist = (int*)(ws + 16640);    // 2048 i32
  float* slotw   = (float*)(ws + 24832);  // 2048 f32
  float* act_rt  = (float*)(ws + 33280);            // 2048*512 f32 = 4 MB
  float* act_sh  = (float*)(ws + 33280 + 4194304);  // 1024*512 f32 = 2 MB

  hipMemsetAsync(ws, 0, 128, stream);  // counts + offsets

  // routing
  gate_kernel<<<TT, 256, 0, stream>>>(X, GW, counts, eids, ew);
  scan_kernel<<<1, 32, 0, stream>>>(counts, offsets);
  fill_kernel<<<(TT + 255) / 256, 256, 0, stream>>>(eids, ew, offsets, toklist, slotw);

  // shared expert: act then down (down writes the full output)
  act_kernel<<<dim3(TT / 16, MM / 128, 1), 128, 0, stream>>>(
      X, WSg, WSu, nullptr, nullptr, nullptr, act_sh, 1);
  down_kernel<<<dim3(TT / 16, HH / 128, 1), 128, 0, stream>>>(
      act_sh, WSd, nullptr, nullptr, nullptr, nullptr, out, 1);

  // routed experts: grouped act then weighted accumulate into out
  act_kernel<<<dim3(TT / 16, MM / 128, EE), 128, 0, stream>>>(
      X, Wg, Wu, toklist, counts, offsets, act_rt, 0);
  down_kernel<<<dim3(TT / 16, HH / 128, EE), 128, 0, stream>>>(
      act_rt, Wd, toklist, slotw, counts, offsets, out, 0);
}